// GCMCEncoder_38792144618255
// MI455X (gfx1250) — compile-verified
//
#include <hip/hip_runtime.h>
#include <hip/hip_bf16.h>

// ---------------------------------------------------------------------------
// GCMC encoder, restructured for MI455X (gfx1250, wave32, WMMA f32 16x16x4):
//   WfP[p,c,{0,1}] = (W_msg@W_out)[2p,c], [2p+1,c]   (fused weights, K-pair interleaved)
//   Y       = X @ Wf                                 (WMMA f32 GEMM, K=128)
//   acc[d] += norm(e) * Y[src(e), r*64 .. r*64+63]   (atomic f32 scatter)
//   out     = relu(acc + bias)
// ---------------------------------------------------------------------------

#define NUM_USERS 100000
#define NUM_ITEMS 50000
#define IN_DIM    128
#define HID_DIM   256
#define OUT_DIM   64
#define R_LEVELS  5
#define E_EDGES   300000
#define WF_COLS   (R_LEVELS * OUT_DIM)   // 320
#define WFP_ROW   (2 * WF_COLS)          // 640 floats per K-pair row
#define EPS_F     1e-10f

typedef __attribute__((ext_vector_type(2))) float v2f;
typedef __attribute__((ext_vector_type(8))) float v8f;

#define LDS_STRIDE 132   // 128 + 4 pad: row stride % 64 banks == 4 -> no bank conflicts

// ---------------------------------------------------------------------------
// Kernel 1: fuse message weight with output weight, writing the K-pair
// interleaved layout consumed directly as WMMA B operands (one b64 per lane):
//   WfP[(k>>1)*640 + c*2 + (k&1)] = sum_h W_msg[r,k,h] * W_out[h,j],  c = r*64+j
// ---------------------------------------------------------------------------
__global__ __launch_bounds__(256) void fuse_weights_kernel(
    const float* __restrict__ Wmsg,   // [R,128,256]
    const float* __restrict__ Wout,   // [256,64]
    float* __restrict__ WfP)          // [64, 320, 2]
{
    int idx = blockIdx.x * blockDim.x + threadIdx.x;
    if (idx >= IN_DIM * WF_COLS) return;
    int k = idx / WF_COLS;
    int c = idx % WF_COLS;
    int r = c / OUT_DIM;
    int j = c % OUT_DIM;
    const float* w1 = Wmsg + ((size_t)r * IN_DIM + k) * HID_DIM;
    float s = 0.0f;
    #pragma unroll 8
    for (int h = 0; h < HID_DIM; ++h)
        s = fmaf(w1[h], Wout[h * OUT_DIM + j], s);
    WfP[(size_t)(k >> 1) * WFP_ROW + c * 2 + (k & 1)] = s;
}

// ---------------------------------------------------------------------------
// Kernel 2: Y[N,320] = X[N,128] @ Wf[128,320], f32 WMMA 16x16x4.
// Block = 160 threads = 5 waves; block owns 16 rows, wave w owns cols
// [64w, 64w+64) (4 WMMA col-tiles). A tile staged in LDS (coalesced), read
// per K-step as ds_load_2addr_b64; B operands are single global_load_b64
// from the pair-interleaved WfP. 128/4 = 32 WMMA steps per accumulator.
// N is always a multiple of 16 here (100000, 50000) -> no bounds checks.
// ---------------------------------------------------------------------------
__global__ __launch_bounds__(160) void wmma_gemm_k128_kernel(
    const float* __restrict__ X,    // [N,128]
    const float* __restrict__ WfP,  // [64,320,2]
    float* __restrict__ Y)          // [N,320]
{
    __shared__ float ldsA[16 * LDS_STRIDE];

    const int tid  = threadIdx.x;
    const int wave = tid >> 5;          // 0..4
    const int lane = tid & 31;
    const int row0 = blockIdx.x * 16;

    // Stage A tile: 16 rows x 128 cols of f32, float4-coalesced.
    for (int idx = tid; idx < 16 * (IN_DIM / 4); idx += 160) {
        int r  = idx >> 5;              // row within tile
        int c4 = idx & 31;              // float4 column
        float4 v = ((const float4*)(X + (size_t)(row0 + r) * IN_DIM))[c4];
        float* dst = &ldsA[r * LDS_STRIDE + c4 * 4];
        dst[0] = v.x; dst[1] = v.y; dst[2] = v.z; dst[3] = v.w;
    }
    __syncthreads();

    const int rA   = lane & 15;         // A row (M) for this lane
    const int kh   = lane >> 4;         // K sub-pair select (0 or 1)
    const int colBase = wave * 64;
    const int nB   = lane & 15;         // B column within 16-wide tile

    v8f acc[4];
    #pragma unroll
    for (int t = 0; t < 4; ++t) acc[t] = (v8f){0,0,0,0,0,0,0,0};

    for (int k0 = 0; k0 < IN_DIM; k0 += 4) {
        // A operand: VGPR0 = K=k0+2*kh, VGPR1 = K=k0+2*kh+1 (ISA 16x4 f32 layout)
        v2f a = *(const v2f*)&ldsA[rA * LDS_STRIDE + k0 + 2 * kh];
        // B operand: the (K=k0+2*kh, K=k0+2*kh+1) pair for column colBase+16t+nB,
        // contiguous in the interleaved layout -> one b64 load per tile.
        const float* wrow = WfP + (size_t)((k0 >> 1) + kh) * WFP_ROW
                                + (size_t)(colBase + nB) * 2;
        #pragma unroll
        for (int t = 0; t < 4; ++t) {
            v2f b = *(const v2f*)&wrow[32 * t];
            acc[t] = __builtin_amdgcn_wmma_f32_16x16x4_f32(
                false, a, false, b, (short)0, acc[t], false, false);
        }
    }

    // Store D: VGPR i -> row (i + 8*(lane>>4)), col = lane&15 within tile.
    const int rowOff = 8 * kh;
    #pragma unroll
    for (int t = 0; t < 4; ++t) {
        float* ybase = Y + (size_t)(row0 + rowOff) * WF_COLS + colBase + 16 * t + nB;
        #pragma unroll
        for (int i = 0; i < 8; ++i)
            ybase[(size_t)i * WF_COLS] = acc[t][i];
    }
}

// ---------------------------------------------------------------------------
// Kernel 3: edge scatter. 16 threads per edge; thread handles one float4
// chunk of the 64-wide fused-output row for the edge's rating level.
//   acc[dst, :] += norm(e) * Y[src, r*64 : r*64+64]
// ---------------------------------------------------------------------------
__global__ __launch_bounds__(256) void edge_scatter_kernel(
    const int*   __restrict__ src_idx,  // [R*E] source rows into Y
    const int*   __restrict__ dst_idx,  // [R*E] destination rows into acc
    const float* __restrict__ deg_src,
    const float* __restrict__ deg_dst,
    const float* __restrict__ Ysrc,     // [Nsrc, 320]
    float*       __restrict__ acc,      // [Ndst, 64]
    int total_edges)                    // R*E
{
    int t = blockIdx.x * blockDim.x + threadIdx.x;
    int e = t >> 4;
    if (e >= total_edges) return;
    int c = t & 15;                     // float4 chunk 0..15
    int r = e / E_EDGES;                // rating level

    int s = src_idx[e];
    int d = dst_idx[e];
    float n = __frsqrt_rn(fmaxf(deg_src[s], EPS_F)) *
              __frsqrt_rn(fmaxf(deg_dst[d], EPS_F));

    float4 y = ((const float4*)(Ysrc + (size_t)s * WF_COLS + r * OUT_DIM))[c];
    float* a = acc + (size_t)d * OUT_DIM + c * 4;
    atomicAdd(a + 0, n * y.x);
    atomicAdd(a + 1, n * y.y);
    atomicAdd(a + 2, n * y.z);
    atomicAdd(a + 3, n * y.w);
}

// ---------------------------------------------------------------------------
// Kernel 4: out = relu(acc + bias), in place on d_out.
// ---------------------------------------------------------------------------
__global__ __launch_bounds__(256) void bias_relu_kernel(
    float* __restrict__ out,
    const float* __restrict__ b_user,
    const float* __restrict__ b_item,
    int n_user_elems, int total)
{
    int i = blockIdx.x * blockDim.x + threadIdx.x;
    if (i >= total) return;
    float b = (i < n_user_elems) ? b_user[i & (OUT_DIM - 1)]
                                 : b_item[i & (OUT_DIM - 1)];
    float v = out[i] + b;
    out[i] = v > 0.0f ? v : 0.0f;
}

// ---------------------------------------------------------------------------
extern "C" void kernel_launch(void* const* d_in, const int* in_sizes, int n_in,
                              void* d_out, int out_size, void* d_ws, size_t ws_size,
                              hipStream_t stream) {
    const int*   u_idx       = (const int*)d_in[0];     // [R,E]
    const int*   v_idx       = (const int*)d_in[1];     // [R,E]
    const float* deg_user    = (const float*)d_in[2];   // [NUM_USERS]
    const float* deg_item    = (const float*)d_in[3];   // [NUM_ITEMS]
    const float* X_u         = (const float*)d_in[4];   // [NUM_USERS,128]
    const float* X_v         = (const float*)d_in[5];   // [NUM_ITEMS,128]
    const float* W_item2user = (const float*)d_in[6];   // [R,128,256]
    const float* W_user2item = (const float*)d_in[7];   // [R,128,256]
    const float* W_out_user  = (const float*)d_in[8];   // [256,64]
    const float* b_out_user  = (const float*)d_in[9];   // [64]
    const float* W_out_item  = (const float*)d_in[10];  // [256,64]
    const float* b_out_item  = (const float*)d_in[11];  // [64]

    float* out   = (float*)d_out;
    float* acc_u = out;                                  // [NUM_USERS,64]
    float* acc_v = out + (size_t)NUM_USERS * OUT_DIM;    // [NUM_ITEMS,64]

    // Workspace layout: WfP_user | WfP_item | Y (reused for Y_v then Y_u)
    char*  ws       = (char*)d_ws;
    float* WfP_user = (float*)ws;                                    // 64*320*2
    float* WfP_item = WfP_user + IN_DIM * WF_COLS;                   // 64*320*2
    float* Y        = (float*)(ws + 2 * IN_DIM * WF_COLS * sizeof(float)); // up to 100000*320

    const int total_edges = R_LEVELS * E_EDGES;          // 1.5M
    const int out_elems   = (NUM_USERS + NUM_ITEMS) * OUT_DIM;

    // Zero the atomic accumulators (d_out is poisoned by the harness).
    hipMemsetAsync(d_out, 0, (size_t)out_elems * sizeof(float), stream);

    // Fuse weights: WfP_user = interleave(W_item2user @ W_out_user); likewise item.
    {
        int n = IN_DIM * WF_COLS;
        int g = (n + 255) / 256;
        fuse_weights_kernel<<<g, 256, 0, stream>>>(W_item2user, W_out_user, WfP_user);
        fuse_weights_kernel<<<g, 256, 0, stream>>>(W_user2item, W_out_item, WfP_item);
    }

    // User pipeline: Y_v = X_v @ Wf_user ; scatter rows v_idx -> acc_u[u_idx].
    wmma_gemm_k128_kernel<<<NUM_ITEMS / 16, 160, 0, stream>>>(X_v, WfP_user, Y);
    {
        int n = total_edges * 16;
        edge_scatter_kernel<<<(n + 255) / 256, 256, 0, stream>>>(
            v_idx, u_idx, deg_item, deg_user, Y, acc_u, total_edges);
    }

    // Item pipeline: Y_u = X_u @ Wf_item ; scatter rows u_idx -> acc_v[v_idx].
    wmma_gemm_k128_kernel<<<NUM_USERS / 16, 160, 0, stream>>>(X_u, WfP_item, Y);
    {
        int n = total_edges * 16;
        edge_scatter_kernel<<<(n + 255) / 256, 256, 0, stream>>>(
            u_idx, v_idx, deg_user, deg_item, Y, acc_v, total_edges);
    }

    // Bias + ReLU in place.
    bias_relu_kernel<<<(out_elems + 255) / 256, 256, 0, stream>>>(
        out, b_out_user, b_out_item, NUM_USERS * OUT_DIM, out_elems);
}